// GNN_cls_32014686224671
// MI455X (gfx1250) — compile-verified
//
#include <hip/hip_runtime.h>
#include <stdint.h>

#define N_NODES   100000
#define N_EDGES   1000000
#define N_GRAPHS  1024
#define F_IN      128
#define N_CLASSES 10
#define BN_EPS    1e-5f

// enc(-inf): bits(-inf)=0xFF800000, sign set -> ~bits
#define NEG_INF_ENC 0x007FFFFFu

typedef __attribute__((ext_vector_type(2))) float v2f;
typedef __attribute__((ext_vector_type(8))) float v8f;

// ---------- order-preserving float <-> uint for atomic max ----------
__device__ __forceinline__ unsigned enc_f32(float f) {
  unsigned u = __float_as_uint(f);
  return (u & 0x80000000u) ? ~u : (u | 0x80000000u);
}
__device__ __forceinline__ float dec_f32(unsigned k) {
  unsigned u = (k & 0x80000000u) ? (k ^ 0x80000000u) : ~k;
  return __uint_as_float(u);
}
__device__ __forceinline__ float elu1(float v) {
  return v > 0.0f ? v : (__expf(v) - 1.0f);
}

// ---------------------------------------------------------------
// Dual GEMM via fp32 WMMA:  T = X @ Wt^T ,  P = X @ Wp^T - X @ Wt^T
// X: [n_rows, K] row-major.  Wt/Wp: [n_out, K] row-major.
// One wave computes one 16x16 output tile for BOTH outputs (shared A frag).
// Requires n_rows % 16 == 0, K % 4 == 0, n_out % 16 == 0 (true here:
// 100000 = 6250*16; K in {128,32,64}; n_out in {32,64}).
// ---------------------------------------------------------------
__global__ void __launch_bounds__(256)
k_gemm_dual_wmma(const float* __restrict__ X, const float* __restrict__ Wt,
                 const float* __restrict__ Wp, float* __restrict__ T,
                 float* __restrict__ P, int K, int n_out,
                 int n_tiles_m, int n_tiles_n)
{
  int wv   = (blockIdx.x * blockDim.x + threadIdx.x) >> 5;  // wave-uniform
  int lane = threadIdx.x & 31;
  int total = n_tiles_m * n_tiles_n;
  if (wv >= total) return;                                  // whole-wave exit
  int mt = wv / n_tiles_n;
  int nt = wv - mt * n_tiles_n;

  int lrow  = lane & 15;          // M (A) / N (B,C,D) index within tile
  int khalf = (lane >> 4) << 1;   // lanes 16-31 hold K+2,K+3

  const float* xr  = X  + (size_t)(mt * 16 + lrow) * K;
  const float* wtr = Wt + (size_t)(nt * 16 + lrow) * K;
  const float* wpr = Wp + (size_t)(nt * 16 + lrow) * K;

  v8f ct = {};   // accumulates X @ Wt^T tile
  v8f cp = {};   // accumulates X @ Wp^T tile
  for (int k = 0; k < K; k += 4) {
    v2f a, bt, bp;
    a.x  = xr [k + khalf];  a.y  = xr [k + khalf + 1];
    bt.x = wtr[k + khalf];  bt.y = wtr[k + khalf + 1];
    bp.x = wpr[k + khalf];  bp.y = wpr[k + khalf + 1];
    ct = __builtin_amdgcn_wmma_f32_16x16x4_f32(false, a, false, bt,
                                               (short)0, ct, false, false);
    cp = __builtin_amdgcn_wmma_f32_16x16x4_f32(false, a, false, bp,
                                               (short)0, cp, false, false);
  }

  // C/D layout: VGPR v, lane L -> C[m][n], m = v + 8*(L>>4), n = L&15
  int nn = nt * 16 + lrow;
  int mb = mt * 16 + ((lane >> 4) << 3);
  #pragma unroll
  for (int v = 0; v < 8; ++v) {
    size_t o = (size_t)(mb + v) * n_out + nn;
    T[o] = ct[v];
    P[o] = cp[v] - ct[v];   // P holds X @ (Wp - Wt)^T
  }
}

// ---------------------------------------------------------------
// Fill uint buffer (sentinel init for segment max)
// ---------------------------------------------------------------
__global__ void k_fill_u32(unsigned* __restrict__ p, unsigned val, int n) {
  int i = blockIdx.x * blockDim.x + threadIdx.x;
  if (i < n) p[i] = val;
}

// ---------------------------------------------------------------
// Per-edge: e = (T[dst]+P[src])*c1 + c0 ; atomic segment-max into U[dst].
// One thread handles one edge x 4 consecutive features (float4 gathers).
// ---------------------------------------------------------------
__global__ void __launch_bounds__(256)
k_edge_max(const float* __restrict__ T, const float* __restrict__ P,
           const int* __restrict__ src, const int* __restrict__ dst,
           const float* __restrict__ c1, const float* __restrict__ c0,
           unsigned* __restrict__ U, int f_out, int total)
{
  int idx = blockIdx.x * blockDim.x + threadIdx.x;
  if (idx >= total) return;
  int f4 = f_out >> 2;
  int e  = idx / f4;
  int c  = (idx - e * f4) << 2;
  int s = src[e];
  int d = dst[e];
  const float4 tv = *reinterpret_cast<const float4*>(T + (size_t)d * f_out + c);
  const float4 pv = *reinterpret_cast<const float4*>(P + (size_t)s * f_out + c);
  unsigned* up = U + (size_t)d * f_out + c;
  atomicMax(up + 0, enc_f32((tv.x + pv.x) * c1[c + 0] + c0[c + 0]));
  atomicMax(up + 1, enc_f32((tv.y + pv.y) * c1[c + 1] + c0[c + 1]));
  atomicMax(up + 2, enc_f32((tv.z + pv.z) * c1[c + 2] + c0[c + 2]));
  atomicMax(up + 3, enc_f32((tv.w + pv.w) * c1[c + 3] + c0[c + 3]));
}

// ---------------------------------------------------------------
// Decode segment-max accumulator; empty segments -> 0; optional ELU.
// ---------------------------------------------------------------
__global__ void k_finalize_elu(const unsigned* __restrict__ U,
                               float* __restrict__ Xo, int n, int do_elu)
{
  int i = blockIdx.x * blockDim.x + threadIdx.x;
  if (i >= n) return;
  unsigned k = U[i];
  float v = (k == NEG_INF_ENC) ? 0.0f : dec_f32(k);
  if (do_elu) v = elu1(v);
  Xo[i] = v;
}

// ---------------------------------------------------------------
// Graph-level max pooling (64 features) via atomic max.
// ---------------------------------------------------------------
__global__ void __launch_bounds__(256)
k_pool_max(const float* __restrict__ X, const int* __restrict__ gid,
           unsigned* __restrict__ GU, int n)
{
  int i = blockIdx.x * blockDim.x + threadIdx.x;
  if (i >= n) return;
  int node = i >> 6;
  int j    = i & 63;
  atomicMax(&GU[((size_t)gid[node] << 6) + j], enc_f32(X[i]));
}

// ---------------------------------------------------------------
// Fold biases + eval-mode BN into per-feature affine (c1, c0):
//   layer1: e' = (T+P+b)*s + (beta - mean*s) ,  s = gamma*rsqrt(var+eps)
//   layer2/3: c1 = 1, c0 = bt+bp
// ---------------------------------------------------------------
__global__ void k_coef(const float* bt1, const float* bp1,
                       const float* gamma, const float* beta,
                       const float* mean, const float* var,
                       const float* bt2, const float* bp2,
                       const float* bt3, const float* bp3,
                       float* c1_1, float* c0_1, float* c1_2, float* c0_2,
                       float* c1_3, float* c0_3)
{
  int j = threadIdx.x;
  if (j < 32) {
    float s = gamma[j] * rsqrtf(var[j] + BN_EPS);
    c1_1[j] = s;
    c0_1[j] = (bt1[j] + bp1[j]) * s + beta[j] - mean[j] * s;
  }
  if (j < 64) {
    c1_2[j] = 1.0f;  c0_2[j] = bt2[j] + bp2[j];
    c1_3[j] = 1.0f;  c0_3[j] = bt3[j] + bp3[j];
  }
}

// ---------------------------------------------------------------
// Small FC: out[r][j] = act(sum_k Xin[r][k]*W[j][k] + b[j])
// ---------------------------------------------------------------
__global__ void k_fc_elu(const float* __restrict__ Xin, const float* __restrict__ W,
                         const float* __restrict__ b, float* __restrict__ Xout,
                         int rows, int fin, int fout, int do_elu)
{
  int i = blockIdx.x * blockDim.x + threadIdx.x;
  if (i >= rows * fout) return;
  int r = i / fout;
  int j = i - r * fout;
  const float* x = Xin + (size_t)r * fin;
  const float* w = W   + (size_t)j * fin;
  float acc = b[j];
  for (int k = 0; k < fin; ++k) acc += x[k] * w[k];
  if (do_elu) acc = elu1(acc);
  Xout[i] = acc;
}

// ---------------------------------------------------------------
// fc3 (32 -> 10) + log_softmax, one thread per graph row.
// ---------------------------------------------------------------
__global__ void k_fc3_logsoftmax(const float* __restrict__ H,
                                 const float* __restrict__ W,
                                 const float* __restrict__ b,
                                 float* __restrict__ out)
{
  int r = blockIdx.x * blockDim.x + threadIdx.x;
  if (r >= N_GRAPHS) return;
  const float* h = H + (size_t)r * 32;
  float l[N_CLASSES];
  float m = -INFINITY;
  #pragma unroll
  for (int j = 0; j < N_CLASSES; ++j) {
    const float* w = W + (size_t)j * 32;
    float a = b[j];
    for (int k = 0; k < 32; ++k) a += h[k] * w[k];
    l[j] = a;
    m = fmaxf(m, a);
  }
  float s = 0.0f;
  #pragma unroll
  for (int j = 0; j < N_CLASSES; ++j) s += expf(l[j] - m);
  float ls = logf(s);
  #pragma unroll
  for (int j = 0; j < N_CLASSES; ++j) out[(size_t)r * N_CLASSES + j] = l[j] - m - ls;
}

// ---------------------------------------------------------------
extern "C" void kernel_launch(void* const* d_in, const int* in_sizes, int n_in,
                              void* d_out, int out_size, void* d_ws, size_t ws_size,
                              hipStream_t stream)
{
  (void)in_sizes; (void)n_in; (void)out_size; (void)ws_size;
  const float* atom = (const float*)d_in[0];
  const int*   esrc = (const int*)d_in[1];
  const int*   edst = (const int*)d_in[2];
  const int*   gid  = (const int*)d_in[3];
  const float* Wt1 = (const float*)d_in[4],  *bt1 = (const float*)d_in[5];
  const float* Wp1 = (const float*)d_in[6],  *bp1 = (const float*)d_in[7];
  const float* bng = (const float*)d_in[8],  *bnb = (const float*)d_in[9];
  const float* bnm = (const float*)d_in[10], *bnv = (const float*)d_in[11];
  const float* Wt2 = (const float*)d_in[12], *bt2 = (const float*)d_in[13];
  const float* Wp2 = (const float*)d_in[14], *bp2 = (const float*)d_in[15];
  const float* Wt3 = (const float*)d_in[16], *bt3 = (const float*)d_in[17];
  const float* Wp3 = (const float*)d_in[18], *bp3 = (const float*)d_in[19];
  const float* Wf1 = (const float*)d_in[20], *bf1 = (const float*)d_in[21];
  const float* Wf2 = (const float*)d_in[22], *bf2 = (const float*)d_in[23];
  const float* Wf3 = (const float*)d_in[24], *bf3 = (const float*)d_in[25];

  // ---- workspace carving (5 big regions of N*64 f32 + small tail) ----
  char* ws = (char*)d_ws;
  const size_t SZ = (size_t)N_NODES * 64 * sizeof(float);  // 25.6 MB
  float*    T  = (float*)(ws);
  float*    P  = (float*)(ws + SZ);
  unsigned* U  = (unsigned*)(ws + 2 * SZ);
  float*    xA = (float*)(ws + 3 * SZ);
  float*    xB = (float*)(ws + 4 * SZ);
  float* c1_1 = (float*)(ws + 5 * SZ);
  float* c0_1 = c1_1 + 32;
  float* c1_2 = c0_1 + 32;
  float* c0_2 = c1_2 + 64;
  float* c1_3 = c0_2 + 64;
  float* c0_3 = c1_3 + 64;
  unsigned* GU = (unsigned*)(c0_3 + 64);
  float* Gf = (float*)(GU + N_GRAPHS * 64);
  float* H1 = Gf + N_GRAPHS * 64;
  float* H2 = H1 + N_GRAPHS * 64;

  k_coef<<<1, 64, 0, stream>>>(bt1, bp1, bng, bnb, bnm, bnv,
                               bt2, bp2, bt3, bp3,
                               c1_1, c0_1, c1_2, c0_2, c1_3, c0_3);

  const int TILES_M = N_NODES / 16;  // 6250, exact

  // ---------------- layer 1: 128 -> 32, BN, seg-max, ELU ----------------
  {
    const int fout = 32, tilesN = fout / 16;
    int waves  = TILES_M * tilesN;
    int blocks = (waves * 32 + 255) / 256;
    k_gemm_dual_wmma<<<blocks, 256, 0, stream>>>(atom, Wt1, Wp1, T, P,
                                                 F_IN, fout, TILES_M, tilesN);
    int n = N_NODES * fout;
    k_fill_u32<<<(n + 255) / 256, 256, 0, stream>>>(U, NEG_INF_ENC, n);
    int tot = N_EDGES * (fout / 4);
    k_edge_max<<<(tot + 255) / 256, 256, 0, stream>>>(T, P, esrc, edst,
                                                      c1_1, c0_1, U, fout, tot);
    k_finalize_elu<<<(n + 255) / 256, 256, 0, stream>>>(U, xA, n, 1);
  }

  // ---------------- layer 2: 32 -> 64, seg-max, ELU ----------------
  {
    const int fin = 32, fout = 64, tilesN = fout / 16;
    int waves  = TILES_M * tilesN;
    int blocks = (waves * 32 + 255) / 256;
    k_gemm_dual_wmma<<<blocks, 256, 0, stream>>>(xA, Wt2, Wp2, T, P,
                                                 fin, fout, TILES_M, tilesN);
    int n = N_NODES * fout;
    k_fill_u32<<<(n + 255) / 256, 256, 0, stream>>>(U, NEG_INF_ENC, n);
    int tot = N_EDGES * (fout / 4);
    k_edge_max<<<(tot + 255) / 256, 256, 0, stream>>>(T, P, esrc, edst,
                                                      c1_2, c0_2, U, fout, tot);
    k_finalize_elu<<<(n + 255) / 256, 256, 0, stream>>>(U, xB, n, 1);
  }

  // ---------------- layer 3: 64 -> 64, seg-max, ELU ----------------
  {
    const int fin = 64, fout = 64, tilesN = fout / 16;
    int waves  = TILES_M * tilesN;
    int blocks = (waves * 32 + 255) / 256;
    k_gemm_dual_wmma<<<blocks, 256, 0, stream>>>(xB, Wt3, Wp3, T, P,
                                                 fin, fout, TILES_M, tilesN);
    int n = N_NODES * fout;
    k_fill_u32<<<(n + 255) / 256, 256, 0, stream>>>(U, NEG_INF_ENC, n);
    int tot = N_EDGES * (fout / 4);
    k_edge_max<<<(tot + 255) / 256, 256, 0, stream>>>(T, P, esrc, edst,
                                                      c1_3, c0_3, U, fout, tot);
    k_finalize_elu<<<(n + 255) / 256, 256, 0, stream>>>(U, xA, n, 1);
  }

  // ---------------- graph max pooling [G,64] ----------------
  {
    int ng = N_GRAPHS * 64;
    k_fill_u32<<<(ng + 255) / 256, 256, 0, stream>>>(GU, NEG_INF_ENC, ng);
    int nn = N_NODES * 64;
    k_pool_max<<<(nn + 255) / 256, 256, 0, stream>>>(xA, gid, GU, nn);
    k_finalize_elu<<<(ng + 255) / 256, 256, 0, stream>>>(GU, Gf, ng, 0);
  }

  // ---------------- MLP head ----------------
  k_fc_elu<<<(N_GRAPHS * 64 + 255) / 256, 256, 0, stream>>>(Gf, Wf1, bf1, H1,
                                                            N_GRAPHS, 64, 64, 1);
  k_fc_elu<<<(N_GRAPHS * 32 + 255) / 256, 256, 0, stream>>>(H1, Wf2, bf2, H2,
                                                            N_GRAPHS, 64, 32, 1);
  k_fc3_logsoftmax<<<(N_GRAPHS + 255) / 256, 256, 0, stream>>>(H2, Wf3, bf3,
                                                               (float*)d_out);
}